// CCERobembedding_69054484185730
// MI455X (gfx1250) — compile-verified
//
#include <hip/hip_runtime.h>

// ROBE hashed-embedding gather for MI455X (gfx1250, wave32).
//
// out[token][c*8+j] = table0[(h0[x[token]][c]+j) & (SIZE-1)]
//                   + table1[(h1[x[token]][c]+j) & (SIZE-1)]
//
// Memory-bound kernel (~0.2 FLOP/byte): optimized for coalesced b128
// traffic and L2 temporal-hint control, not matrix ops.

namespace {

constexpr int kSize    = 262144;      // value-table length, 2^18
constexpr int kMask    = kSize - 1;   // circular gather mask
constexpr int kChunk   = 8;           // floats per hash offset
constexpr int kNChunks = 8;           // offsets per token

// 16-byte vector with only 4-byte alignment guarantee: value-table chunk
// starts are arbitrary float offsets. Clang lowers these to the widest
// legal global_load/store for the target.
typedef float v4f __attribute__((ext_vector_type(4), aligned(4)));

__global__ __launch_bounds__(256) void robe_gather_kernel(
    const float* __restrict__ table0,
    const float* __restrict__ table1,
    const int*   __restrict__ h0,
    const int*   __restrict__ h1,
    const int*   __restrict__ x,
    float*       __restrict__ out,
    int n_tokens)
{
    const int tid   = blockIdx.x * blockDim.x + threadIdx.x;
    const int token = tid >> 3;   // 8 lanes per token -> 4 tokens per wave32
    const int c     = tid & 7;    // chunk index within token
    if (token >= n_tokens) return;

    // Token id: read-once stream (4 MB total) -> non-temporal.
    // 8 consecutive lanes read the same dword (single broadcast fetch).
    const int xi = __builtin_nontemporal_load(&x[token]);

    // Hash rows: 32 B contiguous per token, coalesced across the 8 lanes.
    // These 2x32 MB tables are the thing we WANT resident in the 192 MB L2
    // across graph replays -> default (RT) temporal hint.
    const int o0 = h0[xi * kNChunks + c];
    const int o1 = h1[xi * kNChunks + c];

    float a[kChunk], b[kChunk];

    // Hot path: no wraparound (prob ~3e-5) -> two b128 loads per table.
    // Cold path: masked scalar gather for the circular wrap.
    if (o0 <= kSize - kChunk) {
        const v4f lo = *reinterpret_cast<const v4f*>(table0 + o0);
        const v4f hi = *reinterpret_cast<const v4f*>(table0 + o0 + 4);
        a[0] = lo.x; a[1] = lo.y; a[2] = lo.z; a[3] = lo.w;
        a[4] = hi.x; a[5] = hi.y; a[6] = hi.z; a[7] = hi.w;
    } else {
#pragma unroll
        for (int j = 0; j < kChunk; ++j) a[j] = table0[(o0 + j) & kMask];
    }

    if (o1 <= kSize - kChunk) {
        const v4f lo = *reinterpret_cast<const v4f*>(table1 + o1);
        const v4f hi = *reinterpret_cast<const v4f*>(table1 + o1 + 4);
        b[0] = lo.x; b[1] = lo.y; b[2] = lo.z; b[3] = lo.w;
        b[4] = hi.x; b[5] = hi.y; b[6] = hi.z; b[7] = hi.w;
    } else {
#pragma unroll
        for (int j = 0; j < kChunk; ++j) b[j] = table1[(o1 + j) & kMask];
    }

    v4f r0, r1;
    r0.x = a[0] + b[0]; r0.y = a[1] + b[1];
    r0.z = a[2] + b[2]; r0.w = a[3] + b[3];
    r1.x = a[4] + b[4]; r1.y = a[5] + b[5];
    r1.z = a[6] + b[6]; r1.w = a[7] + b[7];

    // Output: write-once 256 MB stream. Non-temporal so it doesn't evict
    // the hash tables from L2. Lanes write consecutive 32 B blocks ->
    // the wave's two b128 stores together cover 1024 contiguous bytes.
    v4f* dst = reinterpret_cast<v4f*>(out + (size_t)tid * kChunk);
    __builtin_nontemporal_store(r0, dst);
    __builtin_nontemporal_store(r1, dst + 1);
}

} // namespace

extern "C" void kernel_launch(void* const* d_in, const int* in_sizes, int n_in,
                              void* d_out, int out_size, void* d_ws, size_t ws_size,
                              hipStream_t stream) {
    // setup_inputs() order: table0, table1, h0, h1, x
    const float* table0 = (const float*)d_in[0];
    const float* table1 = (const float*)d_in[1];
    const int*   h0     = (const int*)d_in[2];
    const int*   h1     = (const int*)d_in[3];
    const int*   x      = (const int*)d_in[4];
    float*       out    = (float*)d_out;

    const int n_tokens = in_sizes[4];            // N = 1,048,576
    const int total    = n_tokens * kNChunks;    // one thread per (token, chunk)
    const int block    = 256;                    // 8 wave32s per workgroup
    const int grid     = (total + block - 1) / block;

    robe_gather_kernel<<<grid, block, 0, stream>>>(table0, table1, h0, h1, x,
                                                   out, n_tokens);
}